// GraphSAGEModel_91018946937618
// MI455X (gfx1250) — compile-verified
//
#include <hip/hip_runtime.h>

typedef float v2f __attribute__((ext_vector_type(2)));
typedef float v8f __attribute__((ext_vector_type(8)));

// ---------------------------------------------------------------------------
// Zero the accumulators (deg, agg1, agg2). Workspace is poisoned by harness,
// and atomics accumulate into these, so they must be re-zeroed every call.
// ---------------------------------------------------------------------------
__global__ void init_zero_kernel(float* __restrict__ deg,
                                 float* __restrict__ agg1,
                                 float* __restrict__ agg2, int N) {
    int i = blockIdx.x * blockDim.x + threadIdx.x;
    int n16 = N * 16;
    if (i < N)   deg[i]  = 0.0f;
    if (i < n16) { agg1[i] = 0.0f; agg2[i] = 0.0f; }
}

// ---------------------------------------------------------------------------
// C[M x 16] = A[M x K] @ B[K x 16] using V_WMMA_F32_16X16X4_F32 (wave32).
// One wave per 16-row tile of A. K is consumed 4 columns per WMMA.
//
// ISA layouts (CDNA5 ISA):
//   A 16x4 fp32:  VGPR0/1, lanes 0-15 hold M=0..15 @ K={0,1},
//                 lanes 16-31 hold M=0..15 @ K={2,3}
//   B 4x16 fp32:  one row per half-wave per VGPR (K rows striped like A's K)
//   C/D 16x16 f32: VGPR j, lanes 0-15 -> D[j][n], lanes 16-31 -> D[j+8][n]
// Tile guard is wave-uniform so EXEC is all-1s at every WMMA.
// ---------------------------------------------------------------------------
__global__ void gemm_wmma_n16(const float* __restrict__ A,
                              const float* __restrict__ B,
                              float* __restrict__ C, int M, int K) {
    int wavesPerBlock = blockDim.x >> 5;
    int wave = blockIdx.x * wavesPerBlock + ((int)threadIdx.x >> 5);
    int lane = (int)threadIdx.x & 31;
    int tileBase = wave * 16;
    if (tileBase >= M) return;          // uniform across the wave

    int row = lane & 15;                // M within tile (A) / N column (B, C)
    int hi  = lane >> 4;                // half-wave select -> K pair {0,1} vs {2,3}
    int col = row;

    v8f acc = {};
    int nk = K >> 2;
    for (int kk = 0; kk < nk; ++kk) {
        int k0 = (kk << 2) + (hi << 1);

        int ar = tileBase + row;        // clamp ragged tail (reads only)
        if (ar >= M) ar = M - 1;
        v2f a = *(const v2f*)(A + (size_t)ar * K + k0);   // A[ar][k0], A[ar][k0+1]

        v2f b;
        b.x = B[(size_t)k0 * 16 + col];                   // B[k0  ][col]
        b.y = B[(size_t)(k0 + 1) * 16 + col];             // B[k0+1][col]

        acc = __builtin_amdgcn_wmma_f32_16x16x4_f32(
            /*neg_a=*/false, a, /*neg_b=*/false, b,
            /*c_mod=*/(short)0, acc, /*reuse_a=*/false, /*reuse_b=*/false);
    }

#pragma unroll
    for (int j = 0; j < 8; ++j) {
        int r = tileBase + j + (hi << 3);
        if (r < M) C[(size_t)r * 16 + col] = acc[j];
    }
}

// ---------------------------------------------------------------------------
// segment_sum over edges of the 16-wide projected features:
//   agg[dst[e]][0:16] += z[src[e]][0:16]
// 4 threads per edge, each moving one float4 (4 atomic f32 adds). The z row
// and agg row live in L2 (entire problem is L2-resident on MI455X's 192MB L2),
// so this is purely L2 atomic throughput. Degree fused into quarter 0 of the
// first pass (deg != nullptr).
// ---------------------------------------------------------------------------
__global__ void scatter_kernel(const float* __restrict__ z,
                               const int* __restrict__ src,
                               const int* __restrict__ dst,
                               float* __restrict__ agg,
                               float* __restrict__ deg, int E) {
    long long t = (long long)blockIdx.x * blockDim.x + threadIdx.x;
    int e = (int)(t >> 2);
    int q = (int)(t & 3);
    if (e >= E) return;
    int s = src[e];
    int d = dst[e];
    float4 r = ((const float4*)(z + (size_t)s * 16))[q];   // global_load_b128
    float* a = agg + (size_t)d * 16 + q * 4;
    atomicAdd(a + 0, r.x);
    atomicAdd(a + 1, r.y);
    atomicAdd(a + 2, r.z);
    atomicAdd(a + 3, r.w);
    if (deg != nullptr && q == 0) atomicAdd(deg + d, 1.0f);
}

// ---------------------------------------------------------------------------
// out[v][f] = (agg[v][f] + z[v][f]) / (deg[v] + 1) + bias[f]   (+ReLU)
// Valid because segment_sum commutes with @W:  (agg_h + h)/(d+1) @ W
//   = (agg_h@W + h@W)/(d+1) = (agg_z + z)/(d+1).
// ---------------------------------------------------------------------------
__global__ void finish_kernel(const float* __restrict__ agg,
                              const float* __restrict__ z,
                              const float* __restrict__ deg,
                              const float* __restrict__ bias,
                              float* __restrict__ out, int N, int doRelu) {
    int i = blockIdx.x * blockDim.x + threadIdx.x;
    if (i >= N * 16) return;
    int v = i >> 4;
    int f = i & 15;
    float r = (agg[i] + z[i]) / (deg[v] + 1.0f) + bias[f];
    if (doRelu) r = fmaxf(r, 0.0f);
    out[i] = r;
}

// ---------------------------------------------------------------------------
// Inputs (setup_inputs order): features[N,64], W1[64,16], b1[16],
//                              W2[16,16], b2[16], src[E] i32, dst[E] i32
// Output: [N,16] fp32.
// Workspace layout (floats): deg N | z1 16N | agg1 16N | h1 16N | z2 16N |
//                            agg2 16N  -> 81N floats ~= 16.2 MB.
// ---------------------------------------------------------------------------
extern "C" void kernel_launch(void* const* d_in, const int* in_sizes, int n_in,
                              void* d_out, int out_size, void* d_ws, size_t ws_size,
                              hipStream_t stream) {
    const float* features = (const float*)d_in[0];
    const float* W1       = (const float*)d_in[1];
    const float* b1       = (const float*)d_in[2];
    const float* W2       = (const float*)d_in[3];
    const float* b2       = (const float*)d_in[4];
    const int*   src      = (const int*)d_in[5];
    const int*   dst      = (const int*)d_in[6];

    const int N = in_sizes[0] / 64;
    const int E = in_sizes[5];

    float* ws   = (float*)d_ws;
    float* deg  = ws;                        // N
    float* z1   = ws + (size_t)N;            // 16N
    float* agg1 = ws + (size_t)17 * N;       // 16N
    float* h1   = ws + (size_t)33 * N;       // 16N
    float* z2   = ws + (size_t)49 * N;       // 16N
    float* agg2 = ws + (size_t)65 * N;       // 16N

    const int n16 = N * 16;
    const int blk = 256;

    // 0) zero accumulators
    init_zero_kernel<<<(n16 + blk - 1) / blk, blk, 0, stream>>>(deg, agg1, agg2, N);

    // GEMM launch geometry: 4 waves (4 tiles) per 128-thread block
    const int tiles    = (N + 15) / 16;
    const int gemmGrid = (tiles + 3) / 4;

    // 1) z1 = features @ W1            (WMMA fp32, K=64)
    gemm_wmma_n16<<<gemmGrid, 128, 0, stream>>>(features, W1, z1, N, 64);

    // 2) agg1 = segment_sum(z1[src], dst); deg = in-degree
    const long long st = (long long)E * 4;
    scatter_kernel<<<(int)((st + blk - 1) / blk), blk, 0, stream>>>(z1, src, dst, agg1, deg, E);

    // 3) h1 = relu((agg1 + z1)/(deg+1) + b1)
    finish_kernel<<<(n16 + blk - 1) / blk, blk, 0, stream>>>(agg1, z1, deg, b1, h1, N, 1);

    // 4) z2 = h1 @ W2                  (WMMA fp32, K=16)
    gemm_wmma_n16<<<gemmGrid, 128, 0, stream>>>(h1, W2, z2, N, 16);

    // 5) agg2 = segment_sum(z2[src], dst)
    scatter_kernel<<<(int)((st + blk - 1) / blk), blk, 0, stream>>>(z2, src, dst, agg2, nullptr, E);

    // 6) out = (agg2 + z2)/(deg+1) + b2
    finish_kernel<<<(n16 + blk - 1) / blk, blk, 0, stream>>>(agg2, z2, deg, b2, (float*)d_out, N, 0);
}